// HierLogSoftmax_35270271434922
// MI455X (gfx1250) — compile-verified
//
#include <hip/hip_runtime.h>
#include <cstdint>

// ---------------------------------------------------------------------------
// Hierarchical log-softmax, tree: BRANCH=10, DEPTH=3, N=1111 nodes, C=1110.
//
// Per batch row:
//   1) 111 segments of 10: log-softmax (in-place in LDS)
//   2) level-2 nodes (11..110):  lcp[j] += lcp[parent(j)]   (scalar, 100 adds)
//   3) level-3 nodes (111..1110): lcp[c] += lcp[parent(c)]  (WMMA broadcast)
//   4) out[b,0]=0 ; out[b,j]=lcp[j-1]
//
// Bandwidth-bound: ~146 MB total traffic -> ~6.3us at 23.3 TB/s HBM.
// CDNA5 paths: global_load_async_to_lds_b128 for the input tile,
//              v_wmma_f32_16x16x4_f32 for the level-3 parent broadcast.
// ---------------------------------------------------------------------------

typedef __attribute__((ext_vector_type(2))) float v2f;
typedef __attribute__((ext_vector_type(8))) float v8f;

constexpr int TILE_B   = 16;     // batch rows per block (== WMMA M)
constexpr int C_DIM    = 1110;   // scores per row (nodes 1..1110)
constexpr int N_NODES  = 1111;
constexpr int NSEG     = 111;    // internal nodes (segments of 10)
constexpr int NTHREADS = 256;    // 8 wave32

__global__ __launch_bounds__(NTHREADS)
void hier_logsoftmax_kernel(const float* __restrict__ scores,
                            float* __restrict__ out)
{
    // Dynamic LDS: TILE_B * C_DIM floats = 71040 bytes (CDNA5 WGP has 320 KB).
    extern __shared__ float lcp[];   // lcp[r*C_DIM + (node-1)]

    const int tid = threadIdx.x;
    const int row0 = blockIdx.x * TILE_B;
    const float* gsrc = scores + (size_t)row0 * C_DIM;

    // ---- Phase A: async-copy the contiguous 16x1110 f32 tile into LDS ----
    {
        const unsigned lds_base = (unsigned)(uintptr_t)(&lcp[0]); // low 32 bits = LDS offset
        const int nvec = (TILE_B * C_DIM) / 4;                    // 4440 B128 transfers
        for (int v = tid; v < nvec; v += NTHREADS) {
            unsigned lds_addr = lds_base + (unsigned)v * 16u;
            const float* g = gsrc + (size_t)v * 4;
            asm volatile("global_load_async_to_lds_b128 %0, %1, off"
                         :: "v"(lds_addr), "v"(g) : "memory");
        }
        asm volatile("s_wait_asynccnt 0" ::: "memory");
    }
    __syncthreads();

    // ---- Phase B: segmented log-softmax, 16 rows x 111 segments of 10 ----
    for (int task = tid; task < TILE_B * NSEG; task += NTHREADS) {
        const int r = task / NSEG;
        const int s = task - r * NSEG;
        float* seg = &lcp[r * C_DIM + s * 10];
        float m = seg[0];
        #pragma unroll
        for (int i = 1; i < 10; ++i) m = fmaxf(m, seg[i]);
        float sum = 0.f;
        #pragma unroll
        for (int i = 0; i < 10; ++i) sum += __expf(seg[i] - m);
        const float lz = m + __logf(sum);   // x - lz == (x - max) - log(sum exp)
        #pragma unroll
        for (int i = 0; i < 10; ++i) seg[i] -= lz;
    }
    __syncthreads();

    // ---- Phase C1: level-2 accumulate (nodes 11..110, parents 1..10) ----
    for (int task = tid; task < TILE_B * 100; task += NTHREADS) {
        const int r = task / 100;
        const int j = 11 + (task - r * 100);      // node index
        const int p = (j - 1) / 10;               // parent node (level 1)
        lcp[r * C_DIM + j - 1] += lcp[r * C_DIM + p - 1];
    }
    __syncthreads();

    // ---- Phase C2: level-3 accumulate via WMMA parent-broadcast ----
    // For 16 consecutive children c0..c0+15, parents span <=3 consecutive
    // level-2 nodes pmin..pmin+2. D[16x16] = A[16x4] x B[4x16] with
    //   A[m,k] = lp2[row m, node pmin+k]   (f32 A: lanes0-15 M, VGPR0 K0/K2, VGPR1 K1/K3)
    //   B[k,n] = (parent(c0+n) == pmin+k)  (0/1 selection)
    // gives D[m,n] = lp2[m, parent(c0+n)]; add into lcp per the C/D layout.
    {
        const int lane  = tid & 31;
        const int wave  = tid >> 5;
        const int n     = lane & 15;
        const int half  = lane >> 4;        // 0: K=0,1 / M=0..7 ; 1: K=2,3 / M=8..15
        const int kbase = half * 2;
        for (int T = wave; T < 63; T += (NTHREADS / 32)) {
            const int c0   = 111 + 16 * T;          // first child node of tile
            const int pmin = (c0 - 1) / 10;         // its parent (level-2 node)
            // A operand: this lane's two K slices for batch row m = lane&15.
            // Reads of pmin+2/pmin+3 may touch level-3 slots (zero B coeff):
            // values are finite, LDS words are read atomically -> 0*x == 0.
            const int m = n;
            v2f a;
            a.x = lcp[m * C_DIM + (pmin + kbase)     - 1];
            a.y = lcp[m * C_DIM + (pmin + kbase + 1) - 1];
            // B operand: selection matrix column for child c = c0 + n.
            const int c = c0 + n;
            const int p = (c - 1) / 10;
            v2f b;
            b.x = (c <= 1110 && p == pmin + kbase)     ? 1.f : 0.f;
            b.y = (c <= 1110 && p == pmin + kbase + 1) ? 1.f : 0.f;
            v8f acc = {};
            acc = __builtin_amdgcn_wmma_f32_16x16x4_f32(
                      /*neg_a=*/false, a, /*neg_b=*/false, b,
                      /*c_mod=*/(short)0, acc, /*reuse_a=*/false, /*reuse_b=*/false);
            // D layout: VGPR r -> M = r + 8*half, N = lane&15.
            if (c <= 1110) {
                #pragma unroll
                for (int r = 0; r < 8; ++r) {
                    const int mm = r + 8 * half;
                    lcp[mm * C_DIM + c - 1] += acc[r];   // unique (mm, c) writer
                }
            }
        }
    }
    __syncthreads();

    // ---- Phase D: write out [16 x 1111], root column = 0 ----
    float* gout = out + (size_t)row0 * N_NODES;
    for (int idx = tid; idx < TILE_B * N_NODES; idx += NTHREADS) {
        const int r   = idx / N_NODES;
        const int col = idx - r * N_NODES;
        gout[idx] = (col == 0) ? 0.f : lcp[r * C_DIM + col - 1];
    }
}

extern "C" void kernel_launch(void* const* d_in, const int* in_sizes, int n_in,
                              void* d_out, int out_size, void* d_ws, size_t ws_size,
                              hipStream_t stream) {
    (void)n_in; (void)d_ws; (void)ws_size; (void)out_size;
    const float* scores = (const float*)d_in[0];   // [B, 1110] f32
    // d_in[1] ancestor_matrix, d_in[2] seg_ids, d_in[3] child_idx: structure is
    // compile-time known (balanced 10-ary tree), so they are not needed.
    float* out = (float*)d_out;                    // [B, 1111] f32
    const int Bsz  = in_sizes[0] / C_DIM;          // 16384
    const int grid = Bsz / TILE_B;                 // 1024 blocks
    const size_t shmem = (size_t)TILE_B * C_DIM * sizeof(float);  // 71040 B
    hier_logsoftmax_kernel<<<grid, NTHREADS, shmem, stream>>>(scores, out);
}